// GAT1_40510131535940
// MI455X (gfx1250) — compile-verified
//
#include <hip/hip_runtime.h>

#define DIMS 128
#define NEG_SLOPE 0.01f

typedef __attribute__((ext_vector_type(2))) float v2f;
typedef __attribute__((ext_vector_type(8))) float v8f;

// Monotone order-preserving float <-> uint encoding for atomicMax on floats.
__device__ __forceinline__ unsigned enc_f32(float f) {
  unsigned u = __float_as_uint(f);
  return (u & 0x80000000u) ? ~u : (u | 0x80000000u);
}
__device__ __forceinline__ float dec_f32(unsigned v) {
  return __uint_as_float((v & 0x80000000u) ? (v & 0x7fffffffu) : ~v);
}

__device__ __forceinline__ float leaky(float a) {
  return a > 0.f ? a : NEG_SLOPE * a;
}

__global__ void init_seg_kernel(unsigned* __restrict__ mmax, float* __restrict__ esum, int n) {
  int i = blockIdx.x * blockDim.x + threadIdx.x;
  if (i < n) { mmax[i] = 0u; esum[i] = 0.f; }
}

__global__ void zero_f32_kernel(float* __restrict__ p, long long n) {
  long long i = (long long)blockIdx.x * blockDim.x + threadIdx.x;
  if (i < n) p[i] = 0.f;
}

// s_src[n] = h[n] . w_src ; s_dst[n] = h[n] . w_dst  via V_WMMA_F32_16X16X4_F32.
// One wave -> 16 rows of h. A = 16x4 fp32 tile of h, B = 4x16 tile of W where
// W[:,0]=w_src, W[:,1]=w_dst, W[:,2..15]=0. K loop: 128/4 = 32 WMMA issues.
__global__ void score_wmma_kernel(const float* __restrict__ h,
                                  const float* __restrict__ attw_l,  // [2*DIMS]: w_src | w_dst
                                  float* __restrict__ s_src,
                                  float* __restrict__ s_dst, int n) {
  int gwave = (blockIdx.x * blockDim.x + threadIdx.x) >> 5;
  int lane  = threadIdx.x & 31;
  int row0  = gwave * 16;
  if (row0 >= n) return;                 // wave-uniform: EXEC stays all-ones for WMMA

  int m    = lane & 15;                  // A: row index M ; B: col index N
  int half = lane >> 4;                  // K sub-offset selector (K+0/1 vs K+2/3)

  int row = row0 + m;
  if (row >= n) row = n - 1;             // clamp A loads; guarded on writeback
  const float* hrow = h + (long long)row * DIMS;

  // Branchless B operand: lanes N=0/1 read w_src/w_dst, lanes N>=2 read a safe
  // in-bounds address (col m&1) and are zeroed by the lane mask. This keeps the
  // loads unconditional -> no per-step exec save/restore in the inner loop.
  const float bmask = (m < 2) ? 1.f : 0.f;
  const float* wcol = attw_l + (m & 1) * DIMS;

  v8f c = {};
  #pragma unroll
  for (int kk = 0; kk < 32; ++kk) {
    int k0 = kk * 4 + half * 2;
    v2f a; a.x = hrow[k0];          a.y = hrow[k0 + 1];
    v2f b; b.x = bmask * wcol[k0];  b.y = bmask * wcol[k0 + 1];
    // (neg_a, A, neg_b, B, c_mod, C, reuse_a, reuse_b)
    c = __builtin_amdgcn_wmma_f32_16x16x4_f32(false, a, false, b, (short)0, c,
                                              false, false);
  }

  // D layout: VGPR r holds M = r + 8*half, N = lane&15. Only N=0 (s_src) and
  // N=1 (s_dst) columns are meaningful -> lanes 0,16 and 1,17 write out.
  if (m < 2) {
    float* outp = (m == 0) ? s_src : s_dst;
    int mbase = half * 8;
    #pragma unroll
    for (int r = 0; r < 8; ++r) {
      int rr = row0 + mbase + r;
      if (rr < n) outp[rr] = c[r];
    }
  }
}

__global__ void edge_max_kernel(const int* __restrict__ src, const int* __restrict__ dst,
                                const float* __restrict__ s_src, const float* __restrict__ s_dst,
                                unsigned* __restrict__ mmax, int e) {
  int i = blockIdx.x * blockDim.x + threadIdx.x;
  if (i >= e) return;
  int d = dst[i];
  float v = leaky(s_src[src[i]] + s_dst[d]);
  atomicMax(&mmax[d], enc_f32(v));
}

__global__ void edge_sum_kernel(const int* __restrict__ src, const int* __restrict__ dst,
                                const float* __restrict__ s_src, const float* __restrict__ s_dst,
                                const unsigned* __restrict__ mmax, float* __restrict__ esum,
                                int e) {
  int i = blockIdx.x * blockDim.x + threadIdx.x;
  if (i >= e) return;
  int d = dst[i];
  float v = leaky(s_src[src[i]] + s_dst[d]);
  atomicAdd(&esum[d], expf(v - dec_f32(mmax[d])));
}

// One wave per edge: float4-coalesced 512B gather of h[src], alpha recomputed
// from per-node scalars (L2-resident broadcasts), 4 atomicAdds per lane.
__global__ void edge_aggregate_kernel(const int* __restrict__ src, const int* __restrict__ dst,
                                      const float* __restrict__ s_src, const float* __restrict__ s_dst,
                                      const unsigned* __restrict__ mmax, const float* __restrict__ esum,
                                      const float* __restrict__ h_in, float* __restrict__ h_out,
                                      int e) {
  int wid  = (blockIdx.x * blockDim.x + threadIdx.x) >> 5;
  int lane = threadIdx.x & 31;
  if (wid >= e) return;
  int s = src[wid];
  int d = dst[wid];
  float v = leaky(s_src[s] + s_dst[d]);
  float alpha = expf(v - dec_f32(mmax[d])) / esum[d];

  const float4* hp = (const float4*)(h_in + (long long)s * DIMS);
  float4 x = hp[lane];
  float* op = h_out + (long long)d * DIMS + lane * 4;
  atomicAdd(op + 0, x.x * alpha);
  atomicAdd(op + 1, x.y * alpha);
  atomicAdd(op + 2, x.z * alpha);
  atomicAdd(op + 3, x.w * alpha);
}

extern "C" void kernel_launch(void* const* d_in, const int* in_sizes, int n_in,
                              void* d_out, int out_size, void* d_ws, size_t ws_size,
                              hipStream_t stream) {
  const float* h0   = (const float*)d_in[0];
  const int*   src  = (const int*)d_in[1];
  const int*   dst  = (const int*)d_in[2];
  const float* attw = (const float*)d_in[3];

  const int n = in_sizes[0] / DIMS;
  const int e = in_sizes[1];
  const int L = in_sizes[3] / (2 * DIMS);

  // Workspace layout: [h_mid: n*DIMS f32][s_src: n][s_dst: n][mmax: n u32][esum: n]
  char* ws = (char*)d_ws;
  float*    h_mid = (float*)ws;
  float*    ssrc  = (float*)(ws + (size_t)n * DIMS * sizeof(float));
  float*    sdst  = ssrc + n;
  unsigned* mmax  = (unsigned*)(sdst + n);
  float*    esum  = (float*)(mmax + n);

  const long long nd = (long long)n * DIMS;

  for (int l = 0; l < L; ++l) {
    const float* hin  = (l == 0) ? h0 : h_mid;
    float*       hout = (l == L - 1) ? (float*)d_out : h_mid;
    const float* wl   = attw + (size_t)l * 2 * DIMS;

    init_seg_kernel<<<(n + 255) / 256, 256, 0, stream>>>(mmax, esum, n);
    // 4 waves/block, 16 rows/wave -> 64 rows/block
    score_wmma_kernel<<<(n + 63) / 64, 128, 0, stream>>>(hin, wl, ssrc, sdst, n);
    edge_max_kernel<<<(e + 255) / 256, 256, 0, stream>>>(src, dst, ssrc, sdst, mmax, e);
    edge_sum_kernel<<<(e + 255) / 256, 256, 0, stream>>>(src, dst, ssrc, sdst, mmax, esum, e);
    zero_f32_kernel<<<(int)((nd + 255) / 256), 256, 0, stream>>>(hout, nd);
    // 8 edges (waves) per 256-thread block
    edge_aggregate_kernel<<<(e + 7) / 8, 256, 0, stream>>>(src, dst, ssrc, sdst, mmax, esum,
                                                           hin, hout, e);
  }
}